// GNNLDPCDecoder_71614284693535
// MI455X (gfx1250) — compile-verified
//
#include <hip/hip_runtime.h>
#include <cstdint>
#include <cstddef>

// Problem constants (match reference)
#define B_   128
#define B2   (B_ / 2)       // 64 lanes cover the batch with float2
#define N_   20000
#define M_   10
#define NPB  4              // nodes per block in stage kernels
#define TPB  (NPB * B2)     // 256 threads = 8 wave32
#define NEDGE (N_ * M_)

// ---- gfx1250 async global->LDS availability -------------------------------
#if defined(__has_builtin)
#if __has_builtin(__builtin_amdgcn_global_load_async_to_lds_b32)
#define ASYNC_LDS_BUILTIN 1
#endif
#endif
#ifndef ASYNC_LDS_BUILTIN
#define ASYNC_LDS_BUILTIN 0
#endif

typedef __attribute__((address_space(1))) int gas_int;   // global (AS1)
typedef __attribute__((address_space(3))) int las_int;   // LDS (AS3)

__device__ __forceinline__ void async_copy_b32(const int* gsrc, int* ldst) {
#if ASYNC_LDS_BUILTIN
  __builtin_amdgcn_global_load_async_to_lds_b32((gas_int*)gsrc, (las_int*)ldst, 0, 0);
#else
  unsigned lds_off = (unsigned)(uintptr_t)(las_int*)ldst;
  asm volatile("global_load_async_to_lds_b32 %0, %1, %2"
               :: "v"(lds_off), "v"(0u), "s"(gsrc) : "memory");
#endif
}

// Select among 4 edge-type coefficients (used once, in the precompute kernel).
__device__ __forceinline__ float sel4(int e, float a0, float a1, float a2, float a3) {
  float r = (e == 1) ? a1 : a0;
  r = (e == 2) ? a2 : r;
  r = (e == 3) ? a3 : r;
  return r;
}

// Stage this block's idx[] (NPB*M_ ints) and (w,b) coefficient pairs
// (NPB*M_ float2) into LDS via the CDNA5 async global->LDS path (ASYNCcnt).
__device__ __forceinline__ void stage_meta(const int* __restrict__ idx,
                                           const float2* __restrict__ wb,
                                           int ebase, int* s_idx, float2* s_wb) {
  const int t  = threadIdx.x;
  const int ni = NPB * M_;          // 40 index dwords
  const int nw = NPB * M_ * 2;      // 80 coefficient dwords
  if (t < ni + nw) {
    if (t < ni) {
      async_copy_b32(idx + ebase + t, s_idx + t);
    } else {
      const int o = t - ni;
      async_copy_b32((const int*)(wb + ebase) + o, (int*)s_wb + o);
    }
  }
  asm volatile("s_wait_asynccnt 0" ::: "memory");
  __syncthreads();
}

// ---- once: per-edge affine coefficients from edge types --------------------
__global__ void __launch_bounds__(256)
precompute_coeffs(const int* __restrict__ etyp,
                  const float* __restrict__ cew, const float* __restrict__ ceb,
                  const float* __restrict__ vew, const float* __restrict__ veb,
                  float2* __restrict__ cwb, float2* __restrict__ vwb) {
  const int e = blockIdx.x * blockDim.x + threadIdx.x;
  if (e >= NEDGE) return;
  const int tt = etyp[e];
  cwb[e] = make_float2(sel4(tt, cew[0], cew[1], cew[2], cew[3]),
                       sel4(tt, ceb[0], ceb[1], ceb[2], ceb[3]));
  vwb[e] = make_float2(sel4(tt, vew[0], vew[1], vew[2], vew[3]),
                       sel4(tt, veb[0], veb[1], veb[2], veb[3]));
}

// ---- init: transpose input_llr [B,N] -> llr_t [N,B] via LDS tile ----------
__global__ void __launch_bounds__(256)
transpose_BN_to_NB(const float* __restrict__ in, float* __restrict__ out) {
  __shared__ float tile[32][33];
  const int n0 = blockIdx.x * 32;
  const int b0 = blockIdx.y * 32;
  const int tx = threadIdx.x, ty = threadIdx.y;   // 32 x 8
  for (int r = ty; r < 32; r += 8)
    tile[r][tx] = in[(size_t)(b0 + r) * N_ + (n0 + tx)];   // coalesced over n
  __syncthreads();
  for (int r = ty; r < 32; r += 8)
    out[(size_t)(n0 + r) * B_ + (b0 + tx)] = tile[tx][r];  // coalesced over b
}

// ---- check-node update: chk[n,b] = sign_prod * (alpha*min_abs + beta) ------
// Messages in [N, B] layout, viewed as float2 [N, B2]. 64 lanes x 2 elems.
__global__ void __launch_bounds__(TPB)
check_kernel(const float2* __restrict__ var, const int* __restrict__ cidx,
             const float2* __restrict__ cwb, const float* __restrict__ alpha,
             const float* __restrict__ beta, float2* __restrict__ chk) {
  __shared__ int    s_idx[NPB * M_];
  __shared__ float2 s_wb[NPB * M_];
  const int node0 = blockIdx.x * NPB;
  stage_meta(cidx, cwb, node0 * M_, s_idx, s_wb);

  const int t  = threadIdx.x;
  const int ln = t >> 6;            // node within block (0..3)
  const int b2 = t & (B2 - 1);
  const int n  = node0 + ln;
  const float al = alpha[0], be = beta[0];

  unsigned sx0 = 0u, sx1 = 0u;      // xor of sign bits of (x+1e-10)
  float z0 = 1.0f, z1 = 1.0f;       // min |x+1e-10| (0 only if some term == 0)
  float mn0 = 3.0e38f, mn1 = 3.0e38f;
#pragma unroll
  for (int m = 0; m < M_; ++m) {
    const int    id = s_idx[ln * M_ + m];
    const float2 wc = s_wb[ln * M_ + m];
    const int   idm = (id < 0) ? 0 : id;
    const float2 v  = var[idm * B2 + b2];              // coalesced b64 gather
    const float x0 = (id < 0) ? 0.0f : v.x * wc.x + wc.y;
    const float x1 = (id < 0) ? 0.0f : v.y * wc.x + wc.y;
    const float xs0 = x0 + 1e-10f, xs1 = x1 + 1e-10f;
    sx0 ^= __float_as_uint(xs0);  sx1 ^= __float_as_uint(xs1);
    z0 = fminf(z0, fabsf(xs0));   z1 = fminf(z1, fabsf(xs1));
    float a0 = fabsf(x0); a0 = (a0 == 0.0f) ? 1e10f : a0;
    float a1 = fabsf(x1); a1 = (a1 == 0.0f) ? 1e10f : a1;
    mn0 = fminf(mn0, a0);  mn1 = fminf(mn1, a1);
  }
  const float s0 = (z0 == 0.0f) ? 0.0f : ((sx0 & 0x80000000u) ? -1.0f : 1.0f);
  const float s1 = (z1 == 0.0f) ? 0.0f : ((sx1 & 0x80000000u) ? -1.0f : 1.0f);
  chk[n * B2 + b2] = make_float2(s0 * (al * mn0 + be), s1 * (al * mn1 + be));
}

// ---- variable-node update (+ up to 2 residual terms) -----------------------
__global__ void __launch_bounds__(TPB)
var_kernel(const float2* __restrict__ chk, const float2* __restrict__ llr,
           const float2* p1, const float2* p2,
           const int* __restrict__ vidx, const float2* __restrict__ vwb,
           const float* __restrict__ cw, const float* __restrict__ cb,
           const float* __restrict__ wch, const float* __restrict__ wres,
           int nprev, float2* out) {
  __shared__ int    s_idx[NPB * M_];
  __shared__ float2 s_wb[NPB * M_];
  const int node0 = blockIdx.x * NPB;
  stage_meta(vidx, vwb, node0 * M_, s_idx, s_wb);

  const int t  = threadIdx.x;
  const int ln = t >> 6;
  const int b2 = t & (B2 - 1);
  const int n  = node0 + ln;
  const int i  = n * B2 + b2;

  __builtin_prefetch((const float*)(llr + i), 0, 3);   // global_prefetch_b8
  if (nprev > 0) __builtin_prefetch((const float*)(p1 + i), 0, 3);
  if (nprev > 1) __builtin_prefetch((const float*)(p2 + i), 0, 3);

  float s0 = 0.0f, s1 = 0.0f;
#pragma unroll
  for (int m = 0; m < M_; ++m) {
    const int    id = s_idx[ln * M_ + m];
    const float2 wc = s_wb[ln * M_ + m];
    const int   idm = (id < 0) ? 0 : id;
    const float2 g  = chk[idm * B2 + b2];              // coalesced b64 gather
    s0 += (id < 0) ? 0.0f : g.x * wc.x + wc.y;
    s1 += (id < 0) ? 0.0f : g.y * wc.x + wc.y;
  }
  const float W = wch[0], CW = cw[0], CB = cb[0];
  const float2 L = llr[i];
  float v0 = W * L.x + CW * s0 + CB;
  float v1 = W * L.y + CW * s1 + CB;
  if (nprev > 0) { const float2 q = p1[i]; v0 += wres[0] * q.x; v1 += wres[0] * q.y; }
  if (nprev > 1) { const float2 q = p2[i]; v0 += wres[1] * q.x; v1 += wres[1] * q.y; }
  out[i] = make_float2(v0, v1);
}

// ---- final: sigmoid(var + out_w*llr + out_b), transpose [N,B] -> [B,N] -----
__global__ void __launch_bounds__(256)
final_kernel(const float* __restrict__ var, const float* __restrict__ llr,
             const float* __restrict__ ow, const float* __restrict__ ob,
             float* __restrict__ out) {
  __shared__ float tile[32][33];
  const int n0 = blockIdx.x * 32;
  const int b0 = blockIdx.y * 32;
  const float w = ow[0], o = ob[0];
  const int tx = threadIdx.x, ty = threadIdx.y;   // 32 x 8
  for (int r = ty; r < 32; r += 8) {
    const size_t i = (size_t)(n0 + r) * B_ + (b0 + tx);  // coalesced over b
    const float x = var[i] + w * llr[i] + o;
    tile[r][tx] = 1.0f / (1.0f + expf(-x));
  }
  __syncthreads();
  for (int r = ty; r < 32; r += 8)
    out[(size_t)(b0 + r) * N_ + (n0 + tx)] = tile[tx][r]; // coalesced over n
}

// ---------------------------------------------------------------------------
extern "C" void kernel_launch(void* const* d_in, const int* in_sizes, int n_in,
                              void* d_out, int out_size, void* d_ws, size_t ws_size,
                              hipStream_t stream) {
  (void)in_sizes; (void)n_in; (void)out_size; (void)ws_size;

  const float* llr_bn    = (const float*)d_in[0];
  const float* check_ew  = (const float*)d_in[1];
  const float* check_eb  = (const float*)d_in[2];
  const float* alpha     = (const float*)d_in[3];
  const float* beta      = (const float*)d_in[4];
  const float* var_ew    = (const float*)d_in[5];
  const float* var_eb    = (const float*)d_in[6];
  const float* combine_w = (const float*)d_in[7];
  const float* combine_b = (const float*)d_in[8];
  const float* w_ch      = (const float*)d_in[9];
  const float* w_res     = (const float*)d_in[10];
  const float* out_w     = (const float*)d_in[11];
  const float* out_b     = (const float*)d_in[12];
  const int*   cidx      = (const int*)d_in[13];
  const int*   vidx      = (const int*)d_in[14];
  const int*   etyp      = (const int*)d_in[15];
  float*       out       = (float*)d_out;

  const size_t NB = (size_t)N_ * B_;
  float* llr_t = (float*)d_ws;       // [N,B] f32
  float* bufA  = llr_t + NB;
  float* bufB  = bufA + NB;
  float* bufC  = bufB + NB;
  float* chk   = bufC + NB;
  float2* cwb  = (float2*)(chk + NB);          // [N*M] (w,b) for check stage
  float2* vwb  = cwb + NEDGE;                  // [N*M] (w,b) for var stage
  // total ws: 5*NB*4 + 2*NEDGE*8 = 51.2 MB + 3.2 MB

  precompute_coeffs<<<(NEDGE + 255) / 256, 256, 0, stream>>>(
      etyp, check_ew, check_eb, var_ew, var_eb, cwb, vwb);

  const dim3 tb(32, 8);
  const dim3 tg(N_ / 32, B_ / 32);   // 625 x 4
  transpose_BN_to_NB<<<tg, tb, 0, stream>>>(llr_bn, llr_t);

  const int nblk = N_ / NPB;         // 5000 blocks, 256 threads
#define F2(p) ((float2*)(p))
#define CF2(p) ((const float2*)(p))

  // it0: v1 = base(check(v0=llr))
  check_kernel<<<nblk, TPB, 0, stream>>>(CF2(llr_t), cidx, cwb, alpha, beta, F2(chk));
  var_kernel<<<nblk, TPB, 0, stream>>>(CF2(chk), CF2(llr_t), nullptr, nullptr, vidx, vwb,
                                       combine_w, combine_b, w_ch, w_res, 0, F2(bufA));
  // it1: v2 = base(check(v1)) + wr0*v0
  check_kernel<<<nblk, TPB, 0, stream>>>(CF2(bufA), cidx, cwb, alpha, beta, F2(chk));
  var_kernel<<<nblk, TPB, 0, stream>>>(CF2(chk), CF2(llr_t), CF2(llr_t), nullptr, vidx, vwb,
                                       combine_w, combine_b, w_ch, w_res, 1, F2(bufB));
  // it2: v3 = base(check(v2)) + wr0*v1 + wr1*v0
  check_kernel<<<nblk, TPB, 0, stream>>>(CF2(bufB), cidx, cwb, alpha, beta, F2(chk));
  var_kernel<<<nblk, TPB, 0, stream>>>(CF2(chk), CF2(llr_t), CF2(bufA), CF2(llr_t), vidx, vwb,
                                       combine_w, combine_b, w_ch, w_res, 2, F2(bufC));
  // it3: v4 = base(check(v3)) + wr0*v2 + wr1*v1  (same-index in-place over dead v1)
  check_kernel<<<nblk, TPB, 0, stream>>>(CF2(bufC), cidx, cwb, alpha, beta, F2(chk));
  var_kernel<<<nblk, TPB, 0, stream>>>(CF2(chk), CF2(llr_t), CF2(bufB), CF2(bufA), vidx, vwb,
                                       combine_w, combine_b, w_ch, w_res, 2, F2(bufA));
  // it4: v5 = base(check(v4)) + wr0*v3 + wr1*v2  (v4 in bufA already consumed into chk)
  check_kernel<<<nblk, TPB, 0, stream>>>(CF2(bufA), cidx, cwb, alpha, beta, F2(chk));
  var_kernel<<<nblk, TPB, 0, stream>>>(CF2(chk), CF2(llr_t), CF2(bufC), CF2(bufB), vidx, vwb,
                                       combine_w, combine_b, w_ch, w_res, 2, F2(bufA));

  final_kernel<<<tg, tb, 0, stream>>>(bufA, llr_t, out_w, out_b, out);
}